// PCN_Lite_15281493639484
// MI455X (gfx1250) — compile-verified
//
#include <hip/hip_runtime.h>

typedef __attribute__((ext_vector_type(2))) float v2f;
typedef __attribute__((ext_vector_type(8))) float v8f;

#define N_NODES 100000
#define N_EDGES 1600000
#define FIN     64
#define HIDW    128
#define OUTD    10
#define NG      1000

// ---------------------------------------------------------------- utilities
__global__ void zero_f32(float* p, int n) {
  int i = blockIdx.x * blockDim.x + threadIdx.x;
  if (i < n) p[i] = 0.0f;
}

__global__ void deg_kernel(const int* __restrict__ dst, float* __restrict__ deg, int E) {
  int e = blockIdx.x * blockDim.x + threadIdx.x;
  if (e < E) atomicAdd(&deg[dst[e]], 1.0f);
}

__global__ void dinv_kernel(float* deg, int n) {
  int i = blockIdx.x * blockDim.x + threadIdx.x;
  if (i < n) {
    float d = deg[i];
    d = d > 1.0f ? d : 1.0f;
    deg[i] = 1.0f / sqrtf(d);   // deg buffer becomes dinv in place
  }
}

// tmp[n*F+f] = x[n*F+f] * dinv[n]
__global__ void scale_kernel(const float* __restrict__ x, const float* __restrict__ dinv,
                             float* __restrict__ tmp, int total, int F) {
  int i = blockIdx.x * blockDim.x + threadIdx.x;
  if (i < total) tmp[i] = x[i] * dinv[i / F];
}

// agg[dst[e]] += xs[src[e]]  (float4 chunks, L2-resident atomics)
__global__ void spmm_scatter(const float* __restrict__ xs, const int* __restrict__ src,
                             const int* __restrict__ dst, float* __restrict__ agg,
                             int E, int F) {
  int t = blockIdx.x * blockDim.x + threadIdx.x;
  int chunks = F >> 2;               // float4 chunks per row
  int e = t / chunks;
  if (e >= E) return;
  int c = (t - e * chunks) << 2;
  int s = src[e], d = dst[e];
  const float4 v = *reinterpret_cast<const float4*>(xs + (size_t)s * F + c);
  float* ap = agg + (size_t)d * F + c;
  atomicAdd(ap + 0, v.x);
  atomicAdd(ap + 1, v.y);
  atomicAdd(ap + 2, v.z);
  atomicAdd(ap + 3, v.w);
}

// agg[i] = alpha * agg[i] * dinv[n] + beta * X0[i]   (in-place Chebyshev step)
__global__ void cheb_post(float* __restrict__ agg, const float* __restrict__ dinv,
                          const float* __restrict__ X0, float alpha, float beta,
                          int total, int F) {
  int i = blockIdx.x * blockDim.x + threadIdx.x;
  if (i < total) agg[i] = alpha * agg[i] * dinv[i / F] + beta * X0[i];
}

// ------------------------------------------------- fused WMMA GEMM + bias + ReLU
// out[m][n] = relu( sum_b sum_k Xb[m][k] * W[b*F + k][n] + bias[n] )
// One wave (32 lanes) computes one 16x16 tile with V_WMMA_F32_16X16X4_F32.
// Block: (32, 8) -> 8 waves cover the 128 output columns. Grid: M/16 tiles.
__global__ __launch_bounds__(256)
void cheb_gemm_wmma(const float* __restrict__ X0, const float* __restrict__ X1,
                    const float* __restrict__ X2, const float* __restrict__ W,
                    const float* __restrict__ bias, float* __restrict__ out, int F) {
  const int lane = threadIdx.x;            // 0..31, full wave
  const int mt = blockIdx.x * 16;          // row tile (M = 100000 = 6250*16 exact)
  const int nt = threadIdx.y * 16;         // col tile within HID=128
  const int mr   = lane & 15;              // row/col within tile
  const int klo  = (lane >> 4) * 2;        // K sub-offset per A/B layout (16x4 f32)

  v8f acc = {};
  const float* Xs[3] = {X0, X1, X2};
#pragma unroll
  for (int b = 0; b < 3; ++b) {
    const float* __restrict__ X  = Xs[b];
    const float* __restrict__ Wb = W + (size_t)b * F * HIDW;
    const float* __restrict__ Arow = X + (size_t)(mt + mr) * F;
    for (int k0 = 0; k0 < F; k0 += 4) {
      v2f a, bm;
      a.x  = Arow[k0 + klo];
      a.y  = Arow[k0 + klo + 1];
      bm.x = Wb[(size_t)(k0 + klo)     * HIDW + nt + mr];
      bm.y = Wb[(size_t)(k0 + klo + 1) * HIDW + nt + mr];
      // D = A(16x4) * B(4x16) + C, fp32 matrix pipe
      acc = __builtin_amdgcn_wmma_f32_16x16x4_f32(
          /*neg_a=*/false, a, /*neg_b=*/false, bm,
          /*c_mod=*/(short)0, acc, /*reuse_a=*/false, /*reuse_b=*/false);
    }
  }

  // D layout: VGPR i holds M = i (lanes 0-15) / M = i+8 (lanes 16-31), N = lane&15
  const int rbase = mt + ((lane >> 4) << 3);
  const int col   = nt + mr;
  const float bv  = bias[col];
#pragma unroll
  for (int i = 0; i < 8; ++i) {
    float v = acc[i] + bv;
    v = v > 0.0f ? v : 0.0f;
    out[(size_t)(rbase + i) * HIDW + col] = v;
  }
}

// ---------------------------------------------------------------- pooling + FC
__global__ void count_kernel(const int* __restrict__ gid, float* __restrict__ counts, int n) {
  int i = blockIdx.x * blockDim.x + threadIdx.x;
  if (i < n) atomicAdd(&counts[gid[i]], 1.0f);
}

__global__ void pool_kernel(const float* __restrict__ h, const int* __restrict__ gid,
                            float* __restrict__ pooled, int total) {
  int i = blockIdx.x * blockDim.x + threadIdx.x;
  if (i >= total) return;
  int node = i >> 7;           // /HID
  int f    = i & (HIDW - 1);
  atomicAdd(&pooled[(size_t)gid[node] * HIDW + f], h[i]);
}

__global__ void fc_kernel(const float* __restrict__ pooled, const float* __restrict__ counts,
                          const float* __restrict__ Wfc, const float* __restrict__ bfc,
                          float* __restrict__ out) {
  int t = blockIdx.x * blockDim.x + threadIdx.x;
  if (t >= NG * OUTD) return;
  int g = t / OUTD, o = t - g * OUTD;
  float inv = 1.0f / fmaxf(counts[g], 1.0f);
  float s = bfc[o];
  const float* pg = pooled + (size_t)g * HIDW;
#pragma unroll 4
  for (int k = 0; k < HIDW; ++k) s += pg[k] * inv * Wfc[k * OUTD + o];
  out[t] = s;
}

// ---------------------------------------------------------------- launcher
static inline dim3 gblk(long long n) { return dim3((unsigned)((n + 255) / 256)); }

extern "C" void kernel_launch(void* const* d_in, const int* in_sizes, int n_in,
                              void* d_out, int out_size, void* d_ws, size_t ws_size,
                              hipStream_t stream) {
  const float* feat = (const float*)d_in[0];   // N x 64
  const int*   src  = (const int*)d_in[1];     // E
  const int*   dst  = (const int*)d_in[2];     // E
  const int*   gid  = (const int*)d_in[3];     // N
  const float* W1   = (const float*)d_in[4];   // 192 x 128
  const float* b1   = (const float*)d_in[5];   // 128
  const float* W2   = (const float*)d_in[6];   // 384 x 128
  const float* b2   = (const float*)d_in[7];   // 128
  const float* Wfc  = (const float*)d_in[8];   // 128 x 10
  const float* bfc  = (const float*)d_in[9];   // 10
  float* out = (float*)d_out;                  // 1000 x 10

  // workspace layout (floats)
  float* ws    = (float*)d_ws;
  float* dinv  = ws;                                   // N
  float* tmp   = dinv + N_NODES;                       // N*128 (also holds h2)
  float* buf1  = tmp  + (size_t)N_NODES * HIDW;        // N*128 (agg1 -> X1)
  float* buf2  = buf1 + (size_t)N_NODES * HIDW;        // N*128 (agg2 -> X2)
  float* h1    = buf2 + (size_t)N_NODES * HIDW;        // N*128
  float* pooled= h1   + (size_t)N_NODES * HIDW;        // G*128
  float* cnts  = pooled + (size_t)NG * HIDW;           // G

  const dim3 TB(256);

  // ---- degree -> dinv
  zero_f32<<<gblk(N_NODES), TB, 0, stream>>>(dinv, N_NODES);
  deg_kernel<<<gblk(N_EDGES), TB, 0, stream>>>(dst, dinv, N_EDGES);
  dinv_kernel<<<gblk(N_NODES), TB, 0, stream>>>(dinv, N_NODES);

  // ---- layer 1 (F = 64, X0 = feat)
  {
    const int F = FIN;
    const int NF = N_NODES * F;
    scale_kernel<<<gblk(NF), TB, 0, stream>>>(feat, dinv, tmp, NF, F);
    zero_f32<<<gblk(NF), TB, 0, stream>>>(buf1, NF);
    spmm_scatter<<<gblk((long long)N_EDGES * (F / 4)), TB, 0, stream>>>(tmp, src, dst, buf1, N_EDGES, F);
    cheb_post<<<gblk(NF), TB, 0, stream>>>(buf1, dinv, buf1, -1.0f, 0.0f, NF, F);   // X1

    scale_kernel<<<gblk(NF), TB, 0, stream>>>(buf1, dinv, tmp, NF, F);
    zero_f32<<<gblk(NF), TB, 0, stream>>>(buf2, NF);
    spmm_scatter<<<gblk((long long)N_EDGES * (F / 4)), TB, 0, stream>>>(tmp, src, dst, buf2, N_EDGES, F);
    cheb_post<<<gblk(NF), TB, 0, stream>>>(buf2, dinv, feat, -2.0f, -1.0f, NF, F);  // X2

    cheb_gemm_wmma<<<dim3(N_NODES / 16), dim3(32, 8), 0, stream>>>(feat, buf1, buf2, W1, b1, h1, F);
  }

  // ---- layer 2 (F = 128, X0 = h1)
  {
    const int F = HIDW;
    const int NF = N_NODES * F;
    scale_kernel<<<gblk(NF), TB, 0, stream>>>(h1, dinv, tmp, NF, F);
    zero_f32<<<gblk(NF), TB, 0, stream>>>(buf1, NF);
    spmm_scatter<<<gblk((long long)N_EDGES * (F / 4)), TB, 0, stream>>>(tmp, src, dst, buf1, N_EDGES, F);
    cheb_post<<<gblk(NF), TB, 0, stream>>>(buf1, dinv, buf1, -1.0f, 0.0f, NF, F);   // X1

    scale_kernel<<<gblk(NF), TB, 0, stream>>>(buf1, dinv, tmp, NF, F);
    zero_f32<<<gblk(NF), TB, 0, stream>>>(buf2, NF);
    spmm_scatter<<<gblk((long long)N_EDGES * (F / 4)), TB, 0, stream>>>(tmp, src, dst, buf2, N_EDGES, F);
    cheb_post<<<gblk(NF), TB, 0, stream>>>(buf2, dinv, h1, -2.0f, -1.0f, NF, F);    // X2

    cheb_gemm_wmma<<<dim3(N_NODES / 16), dim3(32, 8), 0, stream>>>(h1, buf1, buf2, W2, b2, tmp, F); // h2 -> tmp
  }

  // ---- mean pool per graph + FC
  zero_f32<<<gblk((long long)NG * HIDW), TB, 0, stream>>>(pooled, NG * HIDW);
  zero_f32<<<gblk(NG), TB, 0, stream>>>(cnts, NG);
  count_kernel<<<gblk(N_NODES), TB, 0, stream>>>(gid, cnts, N_NODES);
  pool_kernel<<<gblk((long long)N_NODES * HIDW), TB, 0, stream>>>(tmp, gid, pooled, N_NODES * HIDW);
  fc_kernel<<<gblk((long long)NG * OUTD), TB, 0, stream>>>(pooled, cnts, Wfc, bfc, out);
}